// MemoryBankContrastiveLoss_20229295964475
// MI455X (gfx1250) — compile-verified
//
#include <hip/hip_runtime.h>
#include <math.h>

typedef float v2f __attribute__((ext_vector_type(2)));
typedef float v8f __attribute__((ext_vector_type(8)));

#define C_CLS   19
#define V_VIEWS 10
#define A_DIM   256
#define N_ROWS  190
#define NQ_     38
#define QS_     8192
#define M_COLS  (NQ_ * QS_)            // 311296
#define INV_T   (1.0f / 0.07f)

#define CHUNK   32                     // queue columns per workgroup (divides QS_)
#define NCHUNK  (M_COLS / CHUNK)       // 9728 workgroups in k1; partials per chunk
#define ROWT    12                     // 12 x 16 = 192 padded anchor rows
#define NPAD    192
#define LSTRIDE 260                    // 256 + 4 pad -> conflict-free ds reads

__device__ __forceinline__ float rmax16(float v) {
#pragma unroll
  for (int m = 1; m < 16; m <<= 1) v = fmaxf(v, __shfl_xor(v, m, 16));
  return v;
}
__device__ __forceinline__ float rsum16(float v) {
#pragma unroll
  for (int m = 1; m < 16; m <<= 1) v += __shfl_xor(v, m, 16);
  return v;
}

// ---------------------------------------------------------------------------
// k1: single pass over queue. B tiles staged with GLOBAL_LOAD_ASYNC_TO_LDS_B128
// (ASYNCcnt-tracked DMA, no VGPR round-trip). Per 16x16 output tile
// (K=256 via 64 WMMAs):
//  - flash partials (rowmax, sum exp(l - rowmax)), combined across the two
//    waves through LDS so partials are per-32-col chunk
//  - raw logits of each row's positive class block spilled to posL
// ---------------------------------------------------------------------------
__global__ __launch_bounds__(64) void k1_gemm_stats(
    const float* __restrict__ feat, const int* __restrict__ label,
    const float* __restrict__ queue,
    float* __restrict__ posL, float* __restrict__ pM, float* __restrict__ pS)
{
  __shared__ __attribute__((aligned(16))) float Blds[CHUNK * LSTRIDE]; // 33280 B
  __shared__ __attribute__((aligned(16))) float Alds[16 * LSTRIDE];    // 16640 B
  __shared__ float sM[32], sS[32];          // cross-wave (m,s) combine
  const int chunk = blockIdx.x;
  const int col0  = chunk * CHUNK;
  const int blk   = col0 >> 13;             // queue row index = class block
  const int t     = threadIdx.x;
  const int wave  = t >> 5;
  const int lane  = t & 31;
  const int half  = lane >> 4;
  const int lc    = lane & 15;

  // stage B chunk: 32 queue rows x 256 floats, async DMA direct to LDS
  {
    const float* qbase = queue + (size_t)col0 * A_DIM;
    const unsigned bldsbase = (unsigned)(uintptr_t)Blds;  // flat LDS addr low 32b = LDS offset
#pragma unroll
    for (int i = 0; i < 32; ++i) {
      int linear = i * 64 + t;              // b128 element index 0..2047
      int r  = linear >> 6;
      int f4 = linear & 63;
      unsigned ldsoff = bldsbase + (unsigned)(r * LSTRIDE + f4 * 4) * 4u;
      unsigned goff   = (unsigned)linear * 16u;
      asm volatile("global_load_async_to_lds_b128 %0, %1, %2"
                   :: "v"(ldsoff), "v"(goff), "s"(qbase)
                   : "memory");
    }
    asm volatile("s_wait_asynccnt 0x0" ::: "memory");
  }

  for (int rt = 0; rt < ROWT; ++rt) {
    const int row0 = rt * 16;
    __syncthreads();   // guards Alds + sM/sS reuse; publishes async B on rt==0
    // stage A tile: anchor[n] = feat[n % C][n / C] (view-major flatten), zero-pad
    {
      const float4* f4p = (const float4*)feat;
#pragma unroll
      for (int i = 0; i < 16; ++i) {
        int linear = i * 64 + t;            // float4 index 0..1023
        int r  = linear >> 6;
        int f4 = linear & 63;
        int n  = row0 + r;
        float4 v = make_float4(0.f, 0.f, 0.f, 0.f);
        if (n < N_ROWS) v = f4p[((n % C_CLS) * V_VIEWS + (n / C_CLS)) * 64 + f4];
        float* dst = &Alds[r * LSTRIDE + f4 * 4];
        dst[0] = v.x; dst[1] = v.y; dst[2] = v.z; dst[3] = v.w;
      }
    }
    __syncthreads();

    // 16x16 tile GEMM, K = 256, fp32 WMMA
    v8f acc = {0.f, 0.f, 0.f, 0.f, 0.f, 0.f, 0.f, 0.f};
    const int bcol = wave * 16 + lc;
#pragma unroll 8
    for (int k0 = 0; k0 < A_DIM; k0 += 4) {
      const int kk = k0 + 2 * half;         // ISA 16x4 f32 A layout
      v2f a, b;
      a.x = Alds[lc * LSTRIDE + kk];
      a.y = Alds[lc * LSTRIDE + kk + 1];
      b.x = Blds[bcol * LSTRIDE + kk];
      b.y = Blds[bcol * LSTRIDE + kk + 1];
      acc = __builtin_amdgcn_wmma_f32_16x16x4_f32(false, a, false, b, (short)0,
                                                  acc, false, false);
    }

    // per-row stats for this wave's 16-col tile + positive-block logit spill
    const int posoff = (col0 & (QS_ - 1)) + wave * 16 + lc;
#pragma unroll
    for (int i = 0; i < 8; ++i) {
      const int   mrow = row0 + i + 8 * half;   // C/D layout: M = i + 8*half
      const float li   = acc[i] * INV_T;        // raw logit
      const float mx   = rmax16(li);
      const float se   = rsum16(__expf(li - mx));
      if (mrow < N_ROWS && label[mrow % C_CLS] == blk)
        posL[(size_t)mrow * QS_ + posoff] = li;
      if (lc == 0) {
        sM[wave * 16 + i + 8 * half] = mx;
        sS[wave * 16 + i + 8 * half] = se;
      }
    }
    __syncthreads();
    // combine the two waves' (m,s) -> one partial per chunk per row
    if (t < 16) {
      float m0 = sM[t], s0 = sS[t];
      float m1 = sM[16 + t], s1 = sS[16 + t];
      float nm = fmaxf(m0, m1);
      float ns = s0 * __expf(m0 - nm) + s1 * __expf(m1 - nm);
      pM[(size_t)chunk * NPAD + row0 + t] = nm;
      pS[(size_t)chunk * NPAD + row0 + t] = ns;
    }
  }
}

// k2: flash-combine the 9728 (m,s) partials per row -> rowM, rowS
__global__ __launch_bounds__(256) void k2_combine(
    const float* __restrict__ pM, const float* __restrict__ pS,
    float* __restrict__ rowM, float* __restrict__ rowS)
{
  const int row = blockIdx.x;
  const int t   = threadIdx.x;
  float m = -INFINITY, s = 0.f;
  for (int i = t; i < NCHUNK; i += 256) {
    float pm = pM[(size_t)i * NPAD + row];
    float ps = pS[(size_t)i * NPAD + row];
    float nm = fmaxf(m, pm);
    s = s * __expf(m - nm) + ps * __expf(pm - nm);
    m = nm;
  }
  __shared__ float shM[256], shS[256];
  shM[t] = m; shS[t] = s;
  __syncthreads();
  for (int off = 128; off > 0; off >>= 1) {
    if (t < off) {
      float m2 = shM[t + off], s2 = shS[t + off];
      float nm = fmaxf(shM[t], m2);
      shS[t] = shS[t] * __expf(shM[t] - nm) + s2 * __expf(m2 - nm);
      shM[t] = nm;
    }
    __syncthreads();
  }
  if (t == 0) { rowM[row] = shM[0]; rowS[row] = shS[0]; }
}

// k3: per row, positive-block pass over spilled raw logits (accurate math)
__global__ __launch_bounds__(256) void k3_pos(
    const float* __restrict__ posL, const int* __restrict__ label,
    const float* __restrict__ rowM, const float* __restrict__ rowS,
    float* __restrict__ rowres)
{
  const int n = blockIdx.x;
  const int t = threadIdx.x;
  const int cval = label[n % C_CLS];
  const float M = rowM[n];
  const float S = rowS[n];
  const float* pr = posL + (size_t)n * QS_;
  __shared__ float sh[256];

  float bs = 0.f;
  for (int j = t; j < QS_; j += 256) bs += expf(pr[j] - M);
  sh[t] = bs; __syncthreads();
  for (int off = 128; off > 0; off >>= 1) {
    if (t < off) sh[t] += sh[t + off];
    __syncthreads();
  }
  float neg = S - sh[0];
  if (neg < 0.f) neg = 0.f;
  __syncthreads();

  const int diag = n - cval * QS_;           // self column inside pos block
  float ps = 0.f;
  for (int j = t; j < QS_; j += 256) {
    if (j == diag) continue;
    float l = pr[j] - M;
    ps += l - logf(expf(l) + neg);
  }
  sh[t] = ps; __syncthreads();
  for (int off = 128; off > 0; off >>= 1) {
    if (t < off) sh[t] += sh[t + off];
    __syncthreads();
  }
  if (t == 0) {
    float P = (float)(QS_ - ((diag >= 0 && diag < QS_) ? 1 : 0));
    rowres[n] = sh[0] / P;
  }
}

// k4: loss = -mean(rowres) / BATCH_SIZE
__global__ __launch_bounds__(256) void k4_final(
    const float* __restrict__ rowres, float* __restrict__ out)
{
  const int t = threadIdx.x;
  __shared__ float sh[256];
  sh[t] = (t < N_ROWS) ? rowres[t] : 0.f;
  __syncthreads();
  for (int off = 128; off > 0; off >>= 1) {
    if (t < off) sh[t] += sh[t + off];
    __syncthreads();
  }
  if (t == 0) out[0] = -(sh[0] / (float)N_ROWS) * 0.5f;
}

extern "C" void kernel_launch(void* const* d_in, const int* in_sizes, int n_in,
                              void* d_out, int out_size, void* d_ws, size_t ws_size,
                              hipStream_t stream) {
  (void)in_sizes; (void)n_in; (void)out_size; (void)ws_size;
  const float* feat  = (const float*)d_in[0];
  const int*   label = (const int*)d_in[1];
  const float* queue = (const float*)d_in[2];

  float* ws     = (float*)d_ws;
  float* posL   = ws;                                  // 190*8192  = 1,556,480 f
  float* pM     = posL + (size_t)N_ROWS * QS_;         // 9728*192  = 1,867,776 f
  float* pS     = pM + (size_t)NCHUNK * NPAD;          // 1,867,776 f
  float* rowM   = pS + (size_t)NCHUNK * NPAD;          // 192 f
  float* rowS   = rowM + NPAD;                         // 192 f
  float* rowres = rowS + NPAD;                         // 192 f   (~21.2 MB total)

  k1_gemm_stats<<<NCHUNK, 64, 0, stream>>>(feat, label, queue, posL, pM, pS);
  k2_combine<<<N_ROWS, 256, 0, stream>>>(pM, pS, rowM, rowS);
  k3_pos<<<N_ROWS, 256, 0, stream>>>(posL, label, rowM, rowS, rowres);
  k4_final<<<1, 256, 0, stream>>>(rowres, (float*)d_out);
}